// Loss_45019847197007
// MI455X (gfx1250) — compile-verified
//
#include <hip/hip_runtime.h>

#define N_WORDS   4096
#define N_SAMPLES 2048
#define TPB       256

// 16-byte int vector matching the async-to-LDS builtin's pointee type.
typedef int v4i __attribute__((__vector_size__(16)));

// ---------------------------------------------------------------------------
// Deterministic block reduction: wave32 shuffle tree + LDS across 8 waves.
// Result valid in thread 0. Fixed combination order -> bitwise deterministic.
// ---------------------------------------------------------------------------
__device__ __forceinline__ float block_reduce_sum(float v, float* red) {
    #pragma unroll
    for (int off = 16; off > 0; off >>= 1)
        v += __shfl_down(v, off, 32);
    const int lane = threadIdx.x & 31;
    const int wave = threadIdx.x >> 5;
    if (lane == 0) red[wave] = v;
    __syncthreads();
    if (wave == 0) {
        v = (lane < (TPB >> 5)) ? red[lane] : 0.0f;
        #pragma unroll
        for (int off = 4; off > 0; off >>= 1)  // 8 wave partials
            v += __shfl_down(v, off, 32);
    }
    return v;
}

// ---------------------------------------------------------------------------
// Kernel 1: one block per sample row.
//  - Async-copy the 16KB index row (4096 i32) into LDS (CDNA5 async path).
//  - Each thread gathers 16 bigram[a,b] values (independent loads in flight).
//  - Deterministic block reduction -> ws[s] = inv * per-sample total.
// ---------------------------------------------------------------------------
__global__ void __launch_bounds__(TPB)
sample_gather_kernel(const float* __restrict__ bigram,
                     const float* __restrict__ start,
                     const float* __restrict__ end,
                     const int*   __restrict__ samples,
                     float*       __restrict__ ws_partial) {
    __shared__ int   sIdx[N_WORDS];
    __shared__ float red[TPB >> 5];

    const int s = blockIdx.x;
    const int t = threadIdx.x;
    const int* __restrict__ row = samples + (size_t)s * N_WORDS;

#if defined(__gfx1250__) && __has_builtin(__builtin_amdgcn_global_load_async_to_lds_b128)
    // 4096 ints = 16KB. 4 passes; per pass each of the 256 lanes moves 16B,
    // consecutive lanes -> consecutive 16B chunks (fully coalesced b128s).
    #pragma unroll
    for (int k = 0; k < 4; ++k) {
        const int e = k * (TPB * 4) + t * 4;
        // Step 1: strip const / reinterpret in generic AS; Step 2: addrspace cast.
        v4i* gsrc_generic = (v4i*)(row + e);
        v4i* ldst_generic = (v4i*)(sIdx + e);
        __attribute__((address_space(1))) v4i* gsrc =
            (__attribute__((address_space(1))) v4i*)gsrc_generic;
        __attribute__((address_space(3))) v4i* ldst =
            (__attribute__((address_space(3))) v4i*)ldst_generic;
        __builtin_amdgcn_global_load_async_to_lds_b128(gsrc, ldst,
                                                       /*imm offset*/ 0,
                                                       /*cpol*/ 0);
    }
  #if __has_builtin(__builtin_amdgcn_s_wait_asynccnt)
    __builtin_amdgcn_s_wait_asynccnt(0);
  #else
    asm volatile("s_wait_asynccnt 0" ::: "memory");
  #endif
    __syncthreads();
#else
    // Fallback (also used for the host pass of hipcc): plain b128 loads.
    #pragma unroll
    for (int k = 0; k < 4; ++k) {
        const int e = k * (TPB * 4) + t * 4;
        *(int4*)(sIdx + e) = *(const int4*)(row + e);
    }
    __syncthreads();
#endif

    // 4095 bigram pairs per row; 16 independent gathers per thread.
    float acc = 0.0f;
    #pragma unroll
    for (int j = t; j < N_WORDS - 1; j += TPB) {
        const int a = sIdx[j];
        const int b = sIdx[j + 1];
        acc += bigram[(size_t)a * N_WORDS + b];
    }
    if (t == 0) acc += start[sIdx[0]];
    if (t == 1) acc += end[sIdx[N_WORDS - 1]];

    const float tot = block_reduce_sum(acc, red);
    if (t == 0) ws_partial[s] = tot * (1.0f / (float)N_SAMPLES);
}

// ---------------------------------------------------------------------------
// Kernel 2 (single block): gold diagonal sum + deterministic reduction of the
// 2048 per-sample partials.  out[0] = loss, out[1] = gold_score.
// ---------------------------------------------------------------------------
__global__ void __launch_bounds__(TPB)
finalize_kernel(const float* __restrict__ bigram,
                const float* __restrict__ start,
                const float* __restrict__ end,
                const float* __restrict__ ws_partial,
                float*       __restrict__ out) {
    __shared__ float red[TPB >> 5];
    const int t = threadIdx.x;

    // Gold: sum_i bigram[i, i+1]  (addresses i*4096 + i + 1), i in [0, 4094].
    float diag = 0.0f;
    for (int i = t; i < N_WORDS - 1; i += TPB)
        diag += bigram[(size_t)i * N_WORDS + i + 1];

    // Sample-term total (partials already carry the 1/N_SAMPLES factor).
    float wsum = 0.0f;
    #pragma unroll
    for (int s2 = t; s2 < N_SAMPLES; s2 += TPB)
        wsum += ws_partial[s2];

    const float dtot = block_reduce_sum(diag, red);
    __syncthreads();                      // red[] reused below
    const float stot = block_reduce_sum(wsum, red);

    if (t == 0) {
        const float gold = dtot + start[0] + end[N_WORDS - 1];
        out[0] = stot - gold;             // loss
        out[1] = gold;                    // gold_score
    }
}

// ---------------------------------------------------------------------------
// Host launch: needs 2048 floats (8KB) of d_ws for the per-sample partials.
// Two kernels on the same stream -> serialized; no atomics anywhere.
// ---------------------------------------------------------------------------
extern "C" void kernel_launch(void* const* d_in, const int* in_sizes, int n_in,
                              void* d_out, int out_size, void* d_ws, size_t ws_size,
                              hipStream_t stream) {
    const float* bigram  = (const float*)d_in[0];   // [4096, 4096] f32
    const float* start   = (const float*)d_in[1];   // [4096] f32
    const float* end     = (const float*)d_in[2];   // [4096] f32
    const int*   samples = (const int*)  d_in[3];   // [2048, 4096] i32
    float*       out     = (float*)d_out;           // [loss, gold_score]
    float*       ws      = (float*)d_ws;            // >= 2048 floats

    sample_gather_kernel<<<N_SAMPLES, TPB, 0, stream>>>(bigram, start, end,
                                                        samples, ws);
    finalize_kernel<<<1, TPB, 0, stream>>>(bigram, start, end, ws, out);
}